// GatedAxialAttention_31258771980402
// MI455X (gfx1250) — compile-verified
//
#include <hip/hip_runtime.h>

typedef __attribute__((ext_vector_type(2))) float v2f;
typedef __attribute__((ext_vector_type(8))) float v8f;

#define F_QR 0.3f
#define F_KR 0.3f
#define F_SV 0.5f
#define F_SVE 0.3f
#define BN_EPS 1e-5f

// D = A(16x4) * B(4x16) + C, f32 WMMA (CDNA5 V_WMMA_F32_16X16X4_F32)
__device__ __forceinline__ v8f wmma4(v2f a, v2f b, v8f c) {
  return __builtin_amdgcn_wmma_f32_16x16x4_f32(false, a, false, b, (short)0, c,
                                               false, false);
}

// ---------------------------------------------------------------------------
// Kernel 1: qkv[b] = w_qkv(256x128) @ xb[b](128x128), per-batch GEMM via WMMA.
// Grid 512 (one per b = n*64 + w), 256 threads = 8 waves.
// Wave wv owns M-tiles {2wv, 2wv+1} x all 8 N-tiles -> 16 accumulators.
// ---------------------------------------------------------------------------
__global__ __launch_bounds__(256) void k_qkv(const float* __restrict__ x,
                                             const float* __restrict__ w,
                                             float* __restrict__ qkv) {
  const int b = blockIdx.x;
  const int n = b >> 6, wcol = b & 63;
  const float* xb = x + (size_t)n * (128 * 64 * 128) + (size_t)wcol * 128;
  const int wave = threadIdx.x >> 5;
  const int lane = threadIdx.x & 31;
  const int lh = lane >> 4;   // which K-half of the A/B fragment
  const int lm = lane & 15;   // row/col within tile
  v8f acc[2][8];
#pragma unroll
  for (int t = 0; t < 2; ++t)
#pragma unroll
    for (int no = 0; no < 8; ++no) acc[t][no] = (v8f){};

  for (int k0 = 0; k0 < 128; k0 += 4) {
    const int ka = k0 + 2 * lh;
    v2f af[2];
#pragma unroll
    for (int t = 0; t < 2; ++t) {
      const int m = (wave * 2 + t) * 16 + lm;
      af[t][0] = w[m * 128 + ka];
      af[t][1] = w[m * 128 + ka + 1];
    }
    v2f bf[8];
#pragma unroll
    for (int no = 0; no < 8; ++no) {
      const int nn = no * 16 + lm;
      bf[no][0] = xb[(size_t)ka * 8192 + nn];
      bf[no][1] = xb[(size_t)(ka + 1) * 8192 + nn];
    }
#pragma unroll
    for (int t = 0; t < 2; ++t)
#pragma unroll
      for (int no = 0; no < 8; ++no) acc[t][no] = wmma4(af[t], bf[no], acc[t][no]);
  }

  float* outb = qkv + (size_t)b * (256 * 128);
#pragma unroll
  for (int t = 0; t < 2; ++t)
    for (int no = 0; no < 8; ++no)
#pragma unroll
      for (int r = 0; r < 8; ++r) {
        const int m = (wave * 2 + t) * 16 + lh * 8 + r;
        const int nn = no * 16 + lm;
        outb[m * 128 + nn] = acc[t][no][r];
      }
}

// ---------------------------------------------------------------------------
// Per-channel mean/var over [b][o][h] layout (512 x 256 x 128), block = channel
// ---------------------------------------------------------------------------
__global__ __launch_bounds__(256) void k_chan_stats(const float* __restrict__ src,
                                                    float* __restrict__ meanv,
                                                    float* __restrict__ varv) {
  const int o = blockIdx.x;
  __shared__ float rs[256], rq[256];
  float s = 0.f, q = 0.f;
  for (int i = threadIdx.x; i < 512 * 128; i += 256) {
    const int bb = i >> 7, h = i & 127;
    const float v = src[(size_t)bb * 32768 + o * 128 + h];
    s += v;
    q += v * v;
  }
  rs[threadIdx.x] = s;
  rq[threadIdx.x] = q;
  __syncthreads();
  for (int st = 128; st > 0; st >>= 1) {
    if (threadIdx.x < st) {
      rs[threadIdx.x] += rs[threadIdx.x + st];
      rq[threadIdx.x] += rq[threadIdx.x + st];
    }
    __syncthreads();
  }
  if (threadIdx.x == 0) {
    const float inv = 1.0f / 65536.0f;
    const float m = rs[0] * inv;
    meanv[o] = m;
    varv[o] = rq[0] * inv - m * m;
  }
}

// bn2(bn1(x)) folded to one affine per channel: mean(bn1)=b1, var=g1^2*v/(v+eps)
__global__ void k_affine_qkv(const float* meanv, const float* varv,
                             const float* g1, const float* b1, const float* g2,
                             const float* b2, float* scale, float* shift) {
  const int o = threadIdx.x;
  if (o < 256) {
    const float m = meanv[o], v = varv[o];
    const float r1 = rsqrtf(v + BN_EPS);
    const float v1 = g1[o] * g1[o] * v * r1 * r1;
    const float S = r1 * g1[o] * rsqrtf(v1 + BN_EPS) * g2[o];
    scale[o] = S;
    shift[o] = b2[o] - m * S;
  }
}

// qr[i,j] = F_QR * sum_c q[c,i] * relative[c][i+127-j]
__device__ __forceinline__ float qr_elem(const float* sQ,
                                         const float* __restrict__ rel, int i,
                                         int j) {
  const float* r0 = rel + (i + 127 - j);
  float s = 0.f;
#pragma unroll
  for (int c = 0; c < 8; ++c) s += sQ[c * 128 + i] * r0[c * 255];
  return s * F_QR;
}

// kr[i,j] = F_KR * sum_c k[c,j] * relative[8+c][j+127-i]
__device__ __forceinline__ float kr_elem(const float* sK,
                                         const float* __restrict__ rel, int i,
                                         int j) {
  const float* r0 = rel + 8 * 255 + (j + 127 - i);
  float s = 0.f;
#pragma unroll
  for (int c = 0; c < 8; ++c) s += sK[c * 128 + j] * r0[c * 255];
  return s * F_KR;
}

// ---------------------------------------------------------------------------
// Kernel: global stats of stacked = [qk | qr | kr] per channel (3*8 = 24).
// Grid (512*8) blocks: one per (b, head). qk via WMMA, qr/kr via VALU.
// Values are NOT stored (recomputed in k_attn) -> avoids a 1.6 GB spill.
// ---------------------------------------------------------------------------
__global__ __launch_bounds__(256) void k_stk_stats(
    const float* __restrict__ qkv, const float* __restrict__ scale,
    const float* __restrict__ shift, const float* __restrict__ rel,
    float* __restrict__ ssum, float* __restrict__ ssq) {
  const int b = blockIdx.x >> 3;
  const int g = blockIdx.x & 7;
  __shared__ float sQK[2048];  // q[8][128] then k[8][128], bn applied
  __shared__ float red[256];
  const float* base = qkv + (size_t)b * 32768 + (size_t)g * 4096;
  for (int i = threadIdx.x; i < 2048; i += 256) {
    const int o = g * 32 + (i >> 7);
    sQK[i] = base[i] * scale[o] + shift[o];
  }
  __syncthreads();

  const int wave = threadIdx.x >> 5, lane = threadIdx.x & 31;
  const int lh = lane >> 4, lm = lane & 15;
  float sum_qk = 0.f, sq_qk = 0.f;
#pragma unroll
  for (int no = 0; no < 8; ++no) {
    v8f acc = (v8f){};
#pragma unroll
    for (int k0 = 0; k0 < 8; k0 += 4) {
      const int kc = k0 + 2 * lh;
      v2f a, bb;
      a[0] = sQK[kc * 128 + wave * 16 + lm];         // A[i][c] = q[c][i]
      a[1] = sQK[(kc + 1) * 128 + wave * 16 + lm];
      bb[0] = sQK[1024 + kc * 128 + no * 16 + lm];   // B[c][j] = k[c][j]
      bb[1] = sQK[1024 + (kc + 1) * 128 + no * 16 + lm];
      acc = wmma4(a, bb, acc);
    }
#pragma unroll
    for (int r = 0; r < 8; ++r) {
      const float v = acc[r];
      sum_qk += v;
      sq_qk += v * v;
    }
  }

  float sum_qr = 0.f, sq_qr = 0.f, sum_kr = 0.f, sq_kr = 0.f;
  for (int idx = threadIdx.x; idx < 16384; idx += 256) {
    const int i = idx >> 7, j = idx & 127;
    const float vq = qr_elem(sQK, rel, i, j);
    const float vk = kr_elem(sQK + 1024, rel, i, j);
    sum_qr += vq;
    sq_qr += vq * vq;
    sum_kr += vk;
    sq_kr += vk * vk;
  }

  float vals[6] = {sum_qk, sq_qk, sum_qr, sq_qr, sum_kr, sq_kr};
  for (int t = 0; t < 6; ++t) {
    __syncthreads();
    red[threadIdx.x] = vals[t];
    __syncthreads();
    for (int st = 128; st > 0; st >>= 1) {
      if (threadIdx.x < st) red[threadIdx.x] += red[threadIdx.x + st];
      __syncthreads();
    }
    if (threadIdx.x == 0) {
      const int ch = (t >> 1) * 8 + g;  // 0..7 qk, 8..15 qr, 16..23 kr
      if (t & 1)
        atomicAdd(&ssq[ch], red[0]);
      else
        atomicAdd(&ssum[ch], red[0]);
    }
  }
}

__global__ void k_affine_stk(const float* ssum, const float* ssq,
                             const float* gs, const float* bs, float* sc,
                             float* sh) {
  const int c = threadIdx.x;
  if (c < 24) {
    const float N = 512.0f * 128.0f * 128.0f;
    const float m = ssum[c] / N;
    const float v = ssq[c] / N - m * m;
    const float S = gs[c] * rsqrtf(v + BN_EPS);
    sc[c] = S;
    sh[c] = bs[c] - m * S;
  }
}

// ---------------------------------------------------------------------------
// Fused attention per (b, head): recompute qk (WMMA) + qr/kr, stacked-BN
// affine + sum, softmax over j in LDS (row stride 129 = conflict-free),
// sv via WMMA, sve via VALU.  Writes so[] ALIASED onto qkv[] (this block
// reads exactly the 32 channels it overwrites, all reads before writes).
// ---------------------------------------------------------------------------
__global__ __launch_bounds__(256) void k_attn(
    float* __restrict__ qkv_so, const float* __restrict__ scale,
    const float* __restrict__ shift, const float* __restrict__ rel,
    const float* __restrict__ ssc, const float* __restrict__ ssh) {
  const int b = blockIdx.x >> 3;
  const int g = blockIdx.x & 7;
  __shared__ float sQK[2048];       // q[8][128], k[8][128]
  __shared__ float sVt[128 * 16];   // v transposed: [j][c]
  __shared__ float sL[128 * 129];   // logits / probabilities, padded rows
  float* base = qkv_so + (size_t)b * 32768 + (size_t)g * 4096;

  for (int i = threadIdx.x; i < 2048; i += 256) {
    const int o = g * 32 + (i >> 7);
    sQK[i] = base[i] * scale[o] + shift[o];
  }
  for (int i = threadIdx.x; i < 2048; i += 256) {
    const int c = i >> 7, h = i & 127;
    const int o = g * 32 + 16 + c;
    sVt[h * 16 + c] = base[2048 + i] * scale[o] + shift[o];
  }
  __syncthreads();

  const int wave = threadIdx.x >> 5, lane = threadIdx.x & 31;
  const int lh = lane >> 4, lm = lane & 15;

  // qk -> sL (wave wv owns rows i in [16wv, 16wv+16))
#pragma unroll
  for (int no = 0; no < 8; ++no) {
    v8f acc = (v8f){};
#pragma unroll
    for (int k0 = 0; k0 < 8; k0 += 4) {
      const int kc = k0 + 2 * lh;
      v2f a, bb;
      a[0] = sQK[kc * 128 + wave * 16 + lm];
      a[1] = sQK[(kc + 1) * 128 + wave * 16 + lm];
      bb[0] = sQK[1024 + kc * 128 + no * 16 + lm];
      bb[1] = sQK[1024 + (kc + 1) * 128 + no * 16 + lm];
      acc = wmma4(a, bb, acc);
    }
#pragma unroll
    for (int r = 0; r < 8; ++r) {
      const int i = wave * 16 + lh * 8 + r;
      const int j = no * 16 + lm;
      sL[i * 129 + j] = acc[r];
    }
  }
  __syncthreads();

  // stacked BN affines + sum of the three branches
  const float a_qk = ssc[g], a_qr = ssc[8 + g], a_kr = ssc[16 + g];
  const float csum = ssh[g] + ssh[8 + g] + ssh[16 + g];
  for (int idx = threadIdx.x; idx < 16384; idx += 256) {
    const int i = idx >> 7, j = idx & 127;
    const float vqr = qr_elem(sQK, rel, i, j);
    const float vkr = kr_elem(sQK + 1024, rel, i, j);
    sL[i * 129 + j] = a_qk * sL[i * 129 + j] + a_qr * vqr + a_kr * vkr + csum;
  }
  __syncthreads();

  // softmax over j, one row per thread
  if (threadIdx.x < 128) {
    float* row = sL + threadIdx.x * 129;
    float mx = -3.4e38f;
    for (int j = 0; j < 128; ++j) mx = fmaxf(mx, row[j]);
    float sm = 0.f;
    for (int j = 0; j < 128; ++j) {
      const float e = __expf(row[j] - mx);
      row[j] = e;
      sm += e;
    }
    const float inv = 1.0f / sm;
    for (int j = 0; j < 128; ++j) row[j] *= inv;
  }
  __syncthreads();

  // sv[i][c] = sum_j sim[i][j] * v[c][j] : M=128(i), N=16(c), K=128(j)
  {
    v8f acc = (v8f){};
    for (int k0 = 0; k0 < 128; k0 += 4) {
      const int kc = k0 + 2 * lh;
      v2f a, bb;
      a[0] = sL[(wave * 16 + lm) * 129 + kc];
      a[1] = sL[(wave * 16 + lm) * 129 + kc + 1];
      bb[0] = sVt[kc * 16 + lm];
      bb[1] = sVt[(kc + 1) * 16 + lm];
      acc = wmma4(a, bb, acc);
    }
#pragma unroll
    for (int r = 0; r < 8; ++r) {
      const int i = wave * 16 + lh * 8 + r;
      const int c = lm;
      base[(2 * c) * 128 + i] = F_SV * acc[r];  // so channel g*32 + 2c
    }
  }

  // sve[c][i] = F_SVE * sum_j sim[i][j] * relative[16+c][i+127-j]
  for (int idx = threadIdx.x; idx < 2048; idx += 256) {
    const int c = idx >> 7, i = idx & 127;
    const float* row = sL + i * 129;
    const float* r0 = rel + (16 + c) * 255 + (i + 127);
    float s = 0.f;
    for (int j = 0; j < 128; ++j) s += row[j] * r0[-j];
    base[(2 * c + 1) * 128 + i] = F_SVE * s;  // so channel g*32 + 2c + 1
  }
}

__global__ void k_affine_out(const float* meanv, const float* varv,
                             const float* go, const float* bo, float* sc,
                             float* sh) {
  const int o = threadIdx.x;
  if (o < 256) {
    const float S = go[o] * rsqrtf(varv[o] + BN_EPS);
    sc[o] = S;
    sh[o] = bo[o] - meanv[o] * S;
  }
}

// Final: out[n][oc][w][h] = bn(so)[b][2oc][h] + bn(so)[b][2oc+1][h]
__global__ __launch_bounds__(256) void k_final(const float* __restrict__ so,
                                               const float* __restrict__ osc,
                                               const float* __restrict__ osh,
                                               float* __restrict__ out) {
  const int b = blockIdx.x;
  const int n = b >> 6, wcol = b & 63;
  const float* sob = so + (size_t)b * 32768;
  float* outb = out + (size_t)n * (128 * 64 * 128) + (size_t)wcol * 128;
  for (int idx = threadIdx.x; idx < 128 * 128; idx += 256) {
    const int oc = idx >> 7, h = idx & 127;
    const float y0 = sob[(2 * oc) * 128 + h] * osc[2 * oc] + osh[2 * oc];
    const float y1 = sob[(2 * oc + 1) * 128 + h] * osc[2 * oc + 1] + osh[2 * oc + 1];
    outb[(size_t)oc * 8192 + h] = y0 + y1;
  }
}

extern "C" void kernel_launch(void* const* d_in, const int* in_sizes, int n_in,
                              void* d_out, int out_size, void* d_ws,
                              size_t ws_size, hipStream_t stream) {
  (void)in_sizes; (void)n_in; (void)out_size; (void)ws_size;
  const float* x = (const float*)d_in[0];
  const float* wq = (const float*)d_in[1];
  const float* g1 = (const float*)d_in[2];
  const float* b1 = (const float*)d_in[3];
  const float* g2 = (const float*)d_in[4];
  const float* b2 = (const float*)d_in[5];
  const float* gs = (const float*)d_in[6];
  const float* bs = (const float*)d_in[7];
  const float* go = (const float*)d_in[8];
  const float* bo = (const float*)d_in[9];
  const float* rel = (const float*)d_in[10];
  float* out = (float*)d_out;
  float* ws = (float*)d_ws;

  float* qkv = ws;                  // 16,777,216 floats; later aliased as `so`
  float* st = ws + 16777216;        // small stats block
  float* qm = st;                   // 256
  float* qv = st + 256;             // 256
  float* qsc = st + 512;            // 256
  float* qsh = st + 768;            // 256
  float* ksum = st + 1024;          // 24
  float* ksq = st + 1048;           // 24
  float* ksc = st + 1072;           // 24
  float* ksh = st + 1096;           // 24
  float* om = st + 1120;            // 256
  float* ov = st + 1376;            // 256
  float* osc = st + 1632;           // 256
  float* osh = st + 1888;           // 256 (ends at +2144)

  hipMemsetAsync(ksum, 0, 48 * sizeof(float), stream);  // zero atomic accums

  k_qkv<<<512, 256, 0, stream>>>(x, wq, qkv);
  k_chan_stats<<<256, 256, 0, stream>>>(qkv, qm, qv);
  k_affine_qkv<<<1, 256, 0, stream>>>(qm, qv, g1, b1, g2, b2, qsc, qsh);
  k_stk_stats<<<4096, 256, 0, stream>>>(qkv, qsc, qsh, rel, ksum, ksq);
  k_affine_stk<<<1, 32, 0, stream>>>(ksum, ksq, gs, bs, ksc, ksh);
  k_attn<<<4096, 256, 0, stream>>>(qkv, qsc, qsh, rel, ksc, ksh);
  k_chan_stats<<<256, 256, 0, stream>>>(qkv, om, ov);  // qkv now holds `so`
  k_affine_out<<<1, 256, 0, stream>>>(om, ov, go, bo, osc, osh);
  k_final<<<512, 256, 0, stream>>>(qkv, osc, osh, out);
}